// YOLOXDarkSSL_57346403336570
// MI455X (gfx1250) — compile-verified
//
#include <hip/hip_runtime.h>
#include <math.h>

#define BB 32
#define NN 512
#define DD 1024
#define HH 64
#define WW 64

typedef float v2f __attribute__((ext_vector_type(2)));
typedef float v8f __attribute__((ext_vector_type(8)));

__device__ __forceinline__ float blockReduceSum(float v, float* sm) {
    int t = threadIdx.x;
    sm[t] = v; __syncthreads();
    for (int s = 128; s > 0; s >>= 1) {
        if (t < s) sm[t] += sm[t + s];
        __syncthreads();
    }
    float r = sm[0]; __syncthreads();
    return r;
}

__device__ __forceinline__ float blockReduceMax(float v, float* sm) {
    int t = threadIdx.x;
    sm[t] = v; __syncthreads();
    for (int s = 128; s > 0; s >>= 1) {
        if (t < s) sm[t] = fmaxf(sm[t], sm[t + s]);
        __syncthreads();
    }
    float r = sm[0]; __syncthreads();
    return r;
}

// --- K1: reciprocal row norms for p1 and p2 -------------------------------
__global__ void k_rownorm(const float* __restrict__ p1, const float* __restrict__ p2,
                          float* __restrict__ rn1, float* __restrict__ rn2) {
    __shared__ float sm[256];
    int n = blockIdx.x, b = blockIdx.y, t = threadIdx.x;
    const float* src = blockIdx.z ? p2 : p1;
    float* dst = blockIdx.z ? rn2 : rn1;
    const float* r = src + ((size_t)b * NN + n) * DD;
    float s = 0.f;
#pragma unroll
    for (int j = 0; j < 4; ++j) { float v = r[t + 256 * j]; s += v * v; }
    s = blockReduceSum(s, sm);
    if (t == 0) dst[b * NN + n] = 1.0f / fmaxf(sqrtf(s), 1e-12f);
}

// --- K2: scl term: gather feature vector, cosine sim vs p0 ----------------
__global__ void k_scl(const float* __restrict__ p0, const float* __restrict__ fm,
                      const float* __restrict__ tg, float* __restrict__ s_pn) {
    __shared__ float sm[256];
    int n = blockIdx.x, b = blockIdx.y, t = threadIdx.x;
    float cx = tg[((size_t)b * NN + n) * 5 + 1];
    float cy = tg[((size_t)b * NN + n) * 5 + 2];
    int ix = (int)(cx * ((float)WW / 512.0f) + 0.5f);  // NET_W = 512
    int iy = (int)(cy * ((float)HH / 512.0f) + 0.5f);  // NET_H = 512
    ix = min(max(ix, 0), WW - 1);
    iy = min(max(iy, 0), HH - 1);
    const float* fb = fm + (size_t)b * DD * HH * WW + (size_t)iy * WW + ix;
    const float* pr = p0 + ((size_t)b * NN + n) * DD;
    float d = 0.f, n0 = 0.f, nf = 0.f;
#pragma unroll
    for (int j = 0; j < 4; ++j) {
        int k = t + 256 * j;
        float a = pr[k];
        float f = fb[(size_t)k * HH * WW];
        d += a * f; n0 += a * a; nf += f * f;
    }
    d = blockReduceSum(d, sm);
    n0 = blockReduceSum(n0, sm);
    nf = blockReduceSum(nf, sm);
    if (t == 0)
        s_pn[b * NN + n] = 1.0f - d / (fmaxf(sqrtf(n0), 1e-12f) * fmaxf(sqrtf(nf), 1e-12f));
}

// --- K3: S = (p1 p2^T) * rn1[row] * rn2[col], f32 WMMA 16x16x4 ------------
// 8 waves / block; each wave owns one 16x16 tile of a 32x64 block tile.
// A (16x4 f32): lanes 0-15 row M=l hold K={0,1}; lanes 16-31 hold K={2,3}.
// B (4x16 f32): lanes hold col N=l, same K split (symmetric layout).
__global__ void k_gemm(const float* __restrict__ p1, const float* __restrict__ p2,
                       const float* __restrict__ rn1, const float* __restrict__ rn2,
                       float* __restrict__ S, int c0) {
    int b = c0 + blockIdx.z;
    int lane = threadIdx.x & 31;
    int w = threadIdx.x >> 5;
    int n0 = blockIdx.x * 32 + (w >> 2) * 16;
    int m0 = blockIdx.y * 64 + (w & 3) * 16;
    int l = lane & 15;
    int half = lane >> 4;
    const float* pA = p1 + ((size_t)b * NN + n0 + l) * DD + 2 * half;
    const float* pB = p2 + ((size_t)b * NN + m0 + l) * DD + 2 * half;
    v8f c = {0.f, 0.f, 0.f, 0.f, 0.f, 0.f, 0.f, 0.f};
#pragma unroll 8
    for (int k = 0; k < DD; k += 4) {
        v2f a  = *(const v2f*)(pA + k);
        v2f bv = *(const v2f*)(pB + k);
        c = __builtin_amdgcn_wmma_f32_16x16x4_f32(false, a, false, bv,
                                                  (short)0, c, false, false);
    }
    float cs = rn2[b * NN + m0 + l];
    float* Sb = S + (size_t)blockIdx.z * NN * NN;
#pragma unroll
    for (int v = 0; v < 8; ++v) {
        int row = n0 + v + 8 * half;                 // D layout: VGPR v -> M = v (+8 for hi half)
        Sb[(size_t)row * NN + m0 + l] = c[v] * rn1[b * NN + row] * cs;
    }
}

// --- K4: per-row asym term + diagonal cross-entropy of log_softmax(2*S) ---
__global__ void k_asym_ce(const float* __restrict__ S, float* __restrict__ asym_pn,
                          float* __restrict__ ce_pn, int c0) {
    __shared__ float sm[256];
    int bz = blockIdx.y, b = c0 + bz, n = blockIdx.x, t = threadIdx.x;
    const float* row = S + ((size_t)bz * NN + n) * NN;
    const float* col = S + (size_t)bz * NN * NN + n;
    float x1 = row[t], x2 = row[t + 256];
    float y1 = col[(size_t)t * NN], y2 = col[(size_t)(t + 256) * NN];
    float asum = fabsf(x1 - y1) + fabsf(x2 - y2);
    asum = blockReduceSum(asum, sm);
    float z1 = 2.0f * x1, z2 = 2.0f * x2;            // S / T, T = 0.5
    float mx = blockReduceMax(fmaxf(z1, z2), sm);
    float es = expf(z1 - mx) + expf(z2 - mx);
    es = blockReduceSum(es, sm);
    if (t == 0) {
        asym_pn[b * NN + n] = asum;
        float lse = mx + logf(es);
        ce_pn[b * NN + n] = -(2.0f * row[n] - lse);
    }
}

// --- K5: deterministic batch sums + exact sequential scans ----------------
__global__ void k_final(const float* __restrict__ s_pn, const float* __restrict__ asym_pn,
                        const float* __restrict__ ce_pn, float* __restrict__ out) {
    __shared__ float sm[256];
    __shared__ float sB[BB], aB[BB], cB[BB];
    int t = threadIdx.x;
    for (int b = 0; b < BB; ++b) {
        float v = s_pn[b * NN + t] + s_pn[b * NN + t + 256];
        v = blockReduceSum(v, sm);
        if (t == 0) sB[b] = v;
        v = asym_pn[b * NN + t] + asym_pn[b * NN + t + 256];
        v = blockReduceSum(v, sm);
        if (t == 0) aB[b] = v;
        v = ce_pn[b * NN + t] + ce_pn[b * NN + t + 256];
        v = blockReduceSum(v, sm);
        if (t == 0) cB[b] = v;
    }
    __syncthreads();
    if (t == 0) {
        float scl = 0.f, cyc = 0.f, ssl = 0.f;
        const float cnt = (float)NN * (float)(NN - 1);
        for (int b = 0; b < BB; ++b) {
            scl = (scl + sB[b]) / (float)BB;          // lax.scan (c+s)/B
            cyc = (cyc + aB[b]) / cnt;                // lax.scan (c+r)/cnt
            ssl += (cB[b] / (float)NN) / (float)NN;   // mean diag then /N
        }
        out[0] = scl + cyc + ssl;
    }
}

extern "C" void kernel_launch(void* const* d_in, const int* in_sizes, int n_in,
                              void* d_out, int out_size, void* d_ws, size_t ws_size,
                              hipStream_t stream) {
    const float* p0 = (const float*)d_in[0];
    const float* p1 = (const float*)d_in[1];
    const float* p2 = (const float*)d_in[2];
    const float* fm = (const float*)d_in[3];
    const float* tg = (const float*)d_in[4];
    float* out = (float*)d_out;

    char* ws = (char*)d_ws;
    size_t bn = (size_t)BB * NN;
    float* rn1     = (float*)ws; ws += bn * sizeof(float);
    float* rn2     = (float*)ws; ws += bn * sizeof(float);
    float* s_pn    = (float*)ws; ws += bn * sizeof(float);
    float* asym_pn = (float*)ws; ws += bn * sizeof(float);
    float* ce_pn   = (float*)ws; ws += bn * sizeof(float);

    size_t used = (size_t)(ws - (char*)d_ws);
    size_t al = (used + 255) & ~(size_t)255;
    float* Sbuf = (float*)((char*)d_ws + al);
    size_t perb = (size_t)NN * NN * sizeof(float);    // 1 MB per batch
    size_t avail = ws_size > al ? ws_size - al : 0;
    int Bc = (int)(avail / perb);
    if (Bc > BB) Bc = BB;
    if (Bc < 1) Bc = 1;

    k_rownorm<<<dim3(NN, BB, 2), 256, 0, stream>>>(p1, p2, rn1, rn2);
    k_scl<<<dim3(NN, BB), 256, 0, stream>>>(p0, fm, tg, s_pn);
    for (int c0 = 0; c0 < BB; c0 += Bc) {
        int nb = (BB - c0 < Bc) ? (BB - c0) : Bc;
        k_gemm<<<dim3(NN / 32, NN / 64, nb), 256, 0, stream>>>(p1, p2, rn1, rn2, Sbuf, c0);
        k_asym_ce<<<dim3(NN, nb), 256, 0, stream>>>(Sbuf, asym_pn, ce_pn, c0);
    }
    k_final<<<1, 256, 0, stream>>>(s_pn, asym_pn, ce_pn, out);
}